// CompoundLoss_84482006712841
// MI455X (gfx1250) — compile-verified
//
#include <hip/hip_runtime.h>
#include <cstdint>
#include <cstddef>

// ---------------------------------------------------------------------------
// CompoundLoss (dice + focal + edge) for inputs [8,11,512,512] fp32,
// targets [8,512,512] int32.  Memory-bound single pass (~100MB -> ~4.4us at
// 23.3 TB/s).  CDNA5 paths: TDM tensor_load_to_lds stages the target row-band
// (+halo) into LDS, s_wait_tensorcnt, wave32 shuffle-butterfly reductions.
// v2: per-pixel VALU cut ~2x (packed-field counters via magic multiply,
// sparse one-hot accumulators in private LDS slots via ds_add_f32, focal p_t
// via a cache-hit gather instead of an 11-way select chain) so instruction
// issue no longer exceeds the HBM roofline.
// Deterministic: block partials -> fixed-order tree sum (no float atomics).
// ---------------------------------------------------------------------------

#define NCLS 11
#define IMG_H 512
#define IMG_W 512
#define NBATCH 8
#define ROWS_PER_BLOCK 4
#define BLOCKS_PER_IMG (IMG_H / ROWS_PER_BLOCK) /* 128 */
#define NBLOCKS (NBATCH * BLOCKS_PER_IMG)       /* 1024 */
#define NTHREADS 256
#define NACC 56      /* 5*11 per-class accumulators + focal */
#define SENT 11      /* sentinel class for out-of-image halo cells */
#define NPRIV 22     /* per-thread LDS slots: inter[11] + werr[11] */

typedef unsigned int u32;
typedef u32 uint32x4 __attribute__((ext_vector_type(4)));
typedef int int32x8 __attribute__((ext_vector_type(8)));
typedef int int32x4 __attribute__((ext_vector_type(4)));

__device__ __forceinline__ float wave_sum(float v) {
#pragma unroll
  for (int o = 16; o > 0; o >>= 1) v += __shfl_xor(v, o, 32);
  return v;
}

__global__ __launch_bounds__(NTHREADS) void compound_loss_main(
    const float* __restrict__ logits, const int* __restrict__ tgt,
    float* __restrict__ ws) {
  __shared__ int tl[(ROWS_PER_BLOCK + 2) * IMG_W];   // target rows y0-1..y0+4
  __shared__ float priv[NTHREADS * NPRIV];           // per-thread inter/werr
  __shared__ float red[(NTHREADS / 32) * NACC];

  const int blk = blockIdx.x;
  const int b = blk / BLOCKS_PER_IMG;
  const int y0 = (blk % BLOCKS_PER_IMG) * ROWS_PER_BLOCK;
  const int tid = threadIdx.x;

  // ---- stage target band + halo into LDS via Tensor Data Mover ----
  int cy0 = y0 - 1;
  int ldr0 = 0;
  if (cy0 < 0) { cy0 = 0; ldr0 = 1; }
  int cy1 = y0 + ROWS_PER_BLOCK + 1;
  if (cy1 > IMG_H) cy1 = IMG_H;
  const int ch = cy1 - cy0;  // rows actually loaded (4..6)

  // sentinel-fill halo rows outside the image (disjoint from TDM rows)
  if (ldr0) {
    for (int i = tid; i < IMG_W; i += NTHREADS) tl[i] = SENT;
  }
  if ((y0 + ROWS_PER_BLOCK) == IMG_H) {
    for (int i = tid; i < IMG_W; i += NTHREADS)
      tl[(ROWS_PER_BLOCK + 1) * IMG_W + i] = SENT;
  }
  // zero per-thread private accumulator slots
#pragma unroll
  for (int k = 0; k < NPRIV; k++) priv[k * NTHREADS + tid] = 0.f;

  if (tid < 32) {  // wave 0 issues one TDM load for the whole block
    const u32 lds_off = (u32)(uintptr_t)(const void*)&tl[ldr0 * IMG_W];
    const unsigned long long ga = (unsigned long long)(uintptr_t)(
        tgt + ((size_t)b * IMG_H + (size_t)cy0) * IMG_W);
    uint32x4 g0;
    g0.x = 1u;                                             // count=1 (valid)
    g0.y = lds_off;                                        // lds_addr (bytes)
    g0.z = (u32)ga;                                        // global_addr[31:0]
    g0.w = ((u32)(ga >> 32) & 0x01FFFFFFu) | (2u << 30);   // addr[56:32]|type=2
    int32x8 g1;
    g1[0] = 0x00020000;                 // data_size=2 (4 bytes/elem)
    g1[1] = (IMG_W & 0xFFFF) << 16;     // tensor_dim0[15:0]
    g1[2] = (IMG_H & 0xFFFF) << 16;     // tensor_dim1[15:0] | tensor_dim0[31:16]
    g1[3] = (IMG_W & 0xFFFF) << 16;     // tile_dim0=512    | tensor_dim1[31:16]
    g1[4] = ch;                         // tile_dim1=rows, tile_dim2=0
    g1[5] = IMG_W;                      // tensor_dim0_stride[31:0]
    g1[6] = 0;                          // stride0[47:32] | stride1[15:0]
    g1[7] = (int)(((long long)IMG_W * IMG_H) >> 16);  // stride1[47:16]
    int32x4 g2 = {0, 0, 0, 0};
    int32x4 g3 = {0, 0, 0, 0};
    int32x8 g4 = {0, 0, 0, 0, 0, 0, 0, 0};  // trailing group unused (2D tensor)
    __builtin_amdgcn_tensor_load_to_lds(g0, g1, g2, g3, g4, 0);
    __builtin_amdgcn_s_wait_tensorcnt(0);
  }
  __syncthreads();

  // ---- accumulators ----
  // dense fp32: sum of probs per class (FMA per class per pixel)
  float a_sum[NCLS];
#pragma unroll
  for (int c = 0; c < NCLS; c++) a_sum[c] = 0.f;
  float a_foc = 0.f;
  // packed 6-bit per-class integer counters (max 8 increments/thread < 64):
  // A: classes 0-4 at bit 0,6,12,18,24 ; B: classes 5-9 ; C: class 10
  const u32 MAGIC = 0x01041041u;
  u32 cntA = 0, cntB = 0, cntC = 0;  // onehot pixel counts
  u32 bctA = 0, bctB = 0, bctC = 0;  // boundary counts
  // sparse one-hot accumulators live in private LDS slots (no contention)
  float* myi = &priv[tid];  // slot k at priv[k*NTHREADS + tid]

  const size_t CH = (size_t)IMG_H * IMG_W;  // 1<<18
  const float* pb = logits + (size_t)b * NCLS * CH;
  const float E1 = 0.36787944117144233f;   // exp(-1)
  const float ES2 = 0.24311673443421421f;  // exp(-sqrt(2))

  for (int p = tid; p < ROWS_PER_BLOCK * IMG_W; p += NTHREADS) {
    const int yl = p >> 9;
    const int xc = p & (IMG_W - 1);
    const int gy = y0 + yl;
    const int r0 = yl * IMG_W, r1 = r0 + IMG_W, r2 = r1 + IMG_W;
    const bool okm = (xc > 0), okp = (xc < IMG_W - 1);

    const int t   = tl[r1 + xc];
    const int c00 = okm ? tl[r0 + xc - 1] : SENT;
    const int c01 = tl[r0 + xc];
    const int c02 = okp ? tl[r0 + xc + 1] : SENT;
    const int c10 = okm ? tl[r1 + xc - 1] : SENT;
    const int c12 = okp ? tl[r1 + xc + 1] : SENT;
    const int c20 = okm ? tl[r2 + xc - 1] : SENT;
    const int c21 = tl[r2 + xc];
    const int c22 = okp ? tl[r2 + xc + 1] : SENT;

    // per-cell class bitmasks: 3x3 conv results for all 11 classes at once
    const u32 bC = 1u << t;
    const u32 b00 = 1u << c00, b01 = 1u << c01, b02 = 1u << c02;
    const u32 b10 = 1u << c10, b12 = 1u << c12;
    const u32 b20 = 1u << c20, b21 = 1u << c21, b22 = 1u << c22;
    const u32 any4 = b01 | b10 | b12 | b21;                 // cross (k4)
    const u32 any8 = any4 | b00 | b02 | b20 | b22 | bC;     // 3x3 (k8) >= 1
    const bool allsame = (c00 == t) && (c01 == t) && (c02 == t) &&
                         (c10 == t) && (c12 == t) &&
                         (c20 == t) && (c21 == t) && (c22 == t);
    const u32 all9 = allsame ? bC : 0u;                     // cnt8 == 9
    const u32 bnd = any8 & ~all9;                           // boundary mask

    // packed per-class counts: one magic multiply spreads 5 mask bits into
    // disjoint 6-bit fields (no cross-carries; per-thread count <= 8)
    cntA += (bC & 0x1Fu) * MAGIC;
    cntB += ((bC >> 5) & 0x1Fu) * MAGIC;
    cntC += (bC >> 10) & 1u;
    bctA += (bnd & 0x1Fu) * MAGIC;
    bctB += ((bnd >> 5) & 0x1Fu) * MAGIC;
    bctC += (bnd >> 10) & 1u;

    // 11 coalesced logit streams (class stride = 1<<18 elements)
    const float* px = pb + (size_t)gy * IMG_W + xc;
    float xv[NCLS];
#pragma unroll
    for (int c = 0; c < NCLS; c++) xv[c] = px[(size_t)c * CH];

    float m = xv[0];
    int am = 0;  // argmax, first-max-wins like jnp.argmax
#pragma unroll
    for (int c = 1; c < NCLS; c++) {
      if (xv[c] > m) { m = xv[c]; am = c; }
    }
    float den = 0.f;
#pragma unroll
    for (int c = 0; c < NCLS; c++) { xv[c] = __expf(xv[c] - m); den += xv[c]; }
    const float rden = 1.f / den;
#pragma unroll
    for (int c = 0; c < NCLS; c++) a_sum[c] = fmaf(xv[c], rden, a_sum[c]);

    // p_t via direct gather of x_t (same cachelines as the streams above):
    // replaces an 11-way select chain with 1 cache-hit load + 3 VALU ops
    const float xt = px[(size_t)t * CH];
    const float ptu = __expf(xt - m) * rden;     // unclipped prob at target
    atomicAdd(&myi[t * NTHREADS], ptu);          // inter[t] (private slot)

    const float pt = fminf(fmaxf(ptu, 1e-7f), 1.f);
    const float lp = __logf(pt);
    const float om = 1.f - pt;
    a_foc = fmaf(-0.25f * om * om, lp, a_foc);   // focal term

    // edge werr: |pred_oh - onehot| is nonzero only at classes t and pred
    const bool diff = (am != t);
    const float wt = (diff && ((bnd >> t) & 1u)) ? 1.f : 0.f;          // d=0
    const float wp = (diff && ((bnd >> am) & 1u))
                         ? (((any4 >> am) & 1u) ? E1 : ES2) : 0.f;     // d=1/√2
    atomicAdd(&myi[(11 + t) * NTHREADS], wt);
    atomicAdd(&myi[(11 + am) * NTHREADS], wp);
  }

  // ---- unpack + wave32 butterfly reduction, per-block partial to ws ----
  float a_int[NCLS], a_wer[NCLS], a_cnt[NCLS], a_bct[NCLS];
#pragma unroll
  for (int c = 0; c < NCLS; c++) {
    a_int[c] = myi[c * NTHREADS];
    a_wer[c] = myi[(11 + c) * NTHREADS];
  }
#pragma unroll
  for (int c = 0; c < 5; c++) {
    a_cnt[c] = (float)((cntA >> (6 * c)) & 63u);
    a_cnt[5 + c] = (float)((cntB >> (6 * c)) & 63u);
    a_bct[c] = (float)((bctA >> (6 * c)) & 63u);
    a_bct[5 + c] = (float)((bctB >> (6 * c)) & 63u);
  }
  a_cnt[10] = (float)cntC;
  a_bct[10] = (float)bctC;

#pragma unroll
  for (int c = 0; c < NCLS; c++) {
    a_sum[c] = wave_sum(a_sum[c]);
    a_int[c] = wave_sum(a_int[c]);
    a_cnt[c] = wave_sum(a_cnt[c]);
    a_bct[c] = wave_sum(a_bct[c]);
    a_wer[c] = wave_sum(a_wer[c]);
  }
  a_foc = wave_sum(a_foc);

  const int wave = tid >> 5, lane = tid & 31;
  if (lane == 0) {
    float* r = &red[wave * NACC];
#pragma unroll
    for (int c = 0; c < NCLS; c++) {
      r[c] = a_sum[c];
      r[11 + c] = a_int[c];
      r[22 + c] = a_cnt[c];
      r[33 + c] = a_bct[c];
      r[44 + c] = a_wer[c];
    }
    r[55] = a_foc;
  }
  __syncthreads();
  if (tid < NACC) {
    float s = 0.f;
#pragma unroll
    for (int w = 0; w < NTHREADS / 32; w++) s += red[w * NACC + tid];
    ws[(size_t)blk * NACC + tid] = s;  // deterministic: no atomics
  }
}

__global__ void compound_loss_final(const float* __restrict__ ws,
                                    float* __restrict__ out) {
  __shared__ float S[NBATCH * NACC];  // 448 per-image sums
  const int tid = threadIdx.x;
  for (int i = tid; i < NBATCH * NACC; i += blockDim.x) {
    const int b = i / NACC, j = i % NACC;
    float s = 0.f;
    const float* p = ws + (size_t)(b * BLOCKS_PER_IMG) * NACC + j;
    for (int r = 0; r < BLOCKS_PER_IMG; r++) s += p[(size_t)r * NACC];
    S[i] = s;
  }
  __syncthreads();
  if (tid == 0) {
    const float EPS = 1e-6f;
    // ---- dice ----
    float dice_score = 0.f;
    int nvalid = 0;
    for (int c = 0; c < NCLS; c++) {
      float tot = 0.f, md = 0.f;
      for (int b = 0; b < NBATCH; b++) {
        const float* Sb = &S[b * NACC];
        const float sp = Sb[c], it = Sb[11 + c], cn = Sb[22 + c];
        tot += cn;
        md += (2.f * it + EPS) / (sp + cn + EPS);  // card = sum_p + sum_onehot
      }
      md *= (1.f / (float)NBATCH);
      if (tot > 0.f && c != 0) { nvalid++; dice_score += md; }
    }
    float dice_loss = 0.f;
    if (nvalid > 0) dice_loss = 1.f - dice_score / (float)nvalid;
    // ---- focal ----
    float fs = 0.f;
    for (int b = 0; b < NBATCH; b++) fs += S[b * NACC + 55];
    const float focal_loss = fs / (float)((size_t)NBATCH * IMG_H * IMG_W);
    // ---- edge ----
    float ea = 0.f;
    for (int b = 0; b < NBATCH; b++) {
      const float* Sb = &S[b * NACC];
      float s = 0.f;
      int nv = 0;
      for (int c = 1; c < NCLS; c++) {  // c==0 is IGNORE
        if (Sb[22 + c] > 0.f) {
          nv++;
          s += Sb[44 + c] / fmaxf(Sb[33 + c], 1.f);
        }
      }
      if (nv > 0) ea += s / (float)nv;
    }
    const float edge_loss = ea / (float)NBATCH;

    out[0] = dice_loss + focal_loss + edge_loss;
    out[1] = dice_loss;
    out[2] = focal_loss;
    out[3] = edge_loss;
  }
}

extern "C" void kernel_launch(void* const* d_in, const int* in_sizes, int n_in,
                              void* d_out, int out_size, void* d_ws,
                              size_t ws_size, hipStream_t stream) {
  const float* logits = (const float*)d_in[0];
  const int* tgt = (const int*)d_in[1];
  float* ws = (float*)d_ws;  // needs NBLOCKS*NACC*4 = 224 KB
  float* out = (float*)d_out;
  compound_loss_main<<<NBLOCKS, NTHREADS, 0, stream>>>(logits, tgt, ws);
  compound_loss_final<<<1, 256, 0, stream>>>(ws, out);
}